// Grid4d_HashEncoding_88837103551005
// MI455X (gfx1250) — compile-verified
//
#include <hip/hip_runtime.h>
#include <math.h>

typedef __attribute__((ext_vector_type(2))) float v2f;
typedef __attribute__((ext_vector_type(4))) float v4f;
typedef __attribute__((ext_vector_type(8))) float v8f;

constexpr int kL = 16;
constexpr unsigned kT = 1u << 19;
constexpr int kDFeat = 96;
constexpr int kDHid = 64;
constexpr int kDOut = 16;
constexpr int kBlk = 128;   // points per block == threads per block (4 waves)
constexpr int kFS = 97;     // sF row stride (odd -> conflict-free row access)
constexpr int kW1S = 65;    // sW1 row stride
constexpr int kW2S = 17;    // sW2 row stride

struct ResArr { int r[kL]; };

__global__ __launch_bounds__(kBlk, 4)   // force VGPR<=256: 4 waves/SIMD
void hashgrid_mlp(const float* __restrict__ in4,
                  const float* __restrict__ tab0,
                  const float* __restrict__ tab1,
                  const float* __restrict__ tab2,
                  const float* __restrict__ W1,
                  const float* __restrict__ b1,
                  const float* __restrict__ W2,
                  const float* __restrict__ b2,
                  float* __restrict__ out,
                  int n, ResArr res)
{
  // 49,664 + 24,960 + 4,352 + 320 = 79,296 B  -> 4 blocks / 320KB WGP
  __shared__ float sF[kBlk * kFS];       // feats, then reused for hidden h
  __shared__ float sW1[kDFeat * kW1S];
  __shared__ float sW2[kDHid * kW2S];
  __shared__ float sB1[kDHid];
  __shared__ float sB2[kDOut];

  const int tid = threadIdx.x;

  // stage weights into LDS (row-major with padded stride)
  for (int i = tid; i < kDFeat * kDHid; i += kBlk)
    sW1[(i >> 6) * kW1S + (i & 63)] = W1[i];
  for (int i = tid; i < kDHid * kDOut; i += kBlk)
    sW2[(i >> 4) * kW2S + (i & 15)] = W2[i];
  if (tid < kDHid) sB1[tid] = b1[tid];
  if (tid < kDOut) sB2[tid] = b2[tid];

  // ---------------- Phase A: hash-grid encode (1 point / thread) -----------
  const int p = blockIdx.x * kBlk + tid;
  float x0 = 0.f, x1 = 0.f, x2 = 0.f, x3 = 0.f;
  if (p < n) {
    v4f xin = *(const v4f*)(in4 + (size_t)p * 4);
    x0 = xin.x; x1 = xin.y; x2 = xin.z; x3 = xin.w;
  }

  for (int plane = 0; plane < 3; ++plane) {
    float ux, uy, uz;
    const float* tab;
    if (plane == 0)      { ux = x0; uy = x1; uz = x3; tab = tab0; }
    else if (plane == 1) { ux = x1; uy = x2; uz = x3; tab = tab1; }
    else                 { ux = x0; uy = x2; uz = x3; tab = tab2; }
    ux = (ux + 1.f) * 0.5f; uy = (uy + 1.f) * 0.5f; uz = (uz + 1.f) * 0.5f;

    for (int l = 0; l < kL; ++l) {
      const int rr = res.r[l];
      const float rf = (float)rr;
      const float px = ux * rf, py = uy * rf, pz = uz * rf;
      const float fx = fminf(fmaxf(floorf(px), 0.f), rf - 1.f);
      const float fy = fminf(fmaxf(floorf(py), 0.f), rf - 1.f);
      const float fz = fminf(fmaxf(floorf(pz), 0.f), rf - 1.f);
      const float wx1 = px - fx, wy1 = py - fy, wz1 = pz - fz;
      const float wx0 = 1.f - wx1, wy0 = 1.f - wy1, wz0 = 1.f - wz1;
      const int ix = (int)fx, iy = (int)fy, iz = (int)fz;
      const bool dense =
          ((long long)(rr + 1) * (rr + 1) * (rr + 1) <= (long long)kT);
      const float* tl = tab + (size_t)l * (size_t)kT * 2u;
      float f0 = 0.f, f1 = 0.f;
#pragma unroll
      for (int c = 0; c < 8; ++c) {
        const int cx = ix + (c & 1);
        const int cy = iy + ((c >> 1) & 1);
        const int cz = iz + ((c >> 2) & 1);
        unsigned idx;
        if (dense) {
          idx = (unsigned)(cx + (rr + 1) * (cy + (rr + 1) * cz));
        } else {
          const unsigned h = (unsigned)cx ^ (unsigned)cy * 2654435761u ^
                             (unsigned)cz * 805459861u;
          idx = h & (kT - 1u);
        }
        const float w = ((c & 1) ? wx1 : wx0) * ((c & 2) ? wy1 : wy0) *
                        ((c & 4) ? wz1 : wz0);
        const v2f t = *(const v2f*)(tl + (size_t)idx * 2u);  // global_load_b64
        f0 += w * t.x;
        f1 += w * t.y;
      }
      const int j = plane * 32 + l * 2;
      sF[tid * kFS + j]     = f0;
      sF[tid * kFS + j + 1] = f1;
    }
  }
  __syncthreads();

  // ------------- Coalesced feats writeout (block span is contiguous) -------
  {
    float* gfe = out + (size_t)n * kDOut + (size_t)blockIdx.x * kBlk * kDFeat;
    const int pbase = blockIdx.x * kBlk;
#pragma unroll
    for (int it = 0; it < (kBlk * kDFeat) / (kBlk * 4); ++it) {   // 24 sweeps
      const int f = (it * kBlk + tid) * 4;       // multiple of 4, row len 96
      const int pl = f / kDFeat;
      const int j = f - pl * kDFeat;
      v4f v;
      v.x = sF[pl * kFS + j];
      v.y = sF[pl * kFS + j + 1];
      v.z = sF[pl * kFS + j + 2];
      v.w = sF[pl * kFS + j + 3];
      if (pbase + pl < n) *(v4f*)(gfe + f) = v;  // global_store_b128, coalesced
    }
  }

  // ---------------- Phase B: MLP via V_WMMA_F32_16X16X4_F32 ----------------
  // A 16x4 f32 layout: lanes 0-15 -> M, VGPR0={K0|K2}, VGPR1={K1|K3}
  // B 4x16 f32 layout (mirrored): lanes -> N, VGPR0={K0|K2}, VGPR1={K1|K3}
  // C/D 16x16 f32: VGPR r: lanes0-15 M=r, lanes16-31 M=r+8; N = lane&15
  const int wave = tid >> 5;
  const int lane = tid & 31;
  const int half = lane >> 4;
  const int lr = lane & 15;

  // Layer 1: feats(128x96) @ W1(96x64). Wave owns M-tiles {2w,2w+1} x 4 N-tiles.
  v8f acc[2][4];
#pragma unroll
  for (int tm = 0; tm < 2; ++tm)
#pragma unroll
    for (int nt = 0; nt < 4; ++nt) acc[tm][nt] = (v8f){};

#pragma unroll
  for (int tm = 0; tm < 2; ++tm) {
    const int mt = wave * 2 + tm;
    const float* fb = &sF[(mt * 16 + lr) * kFS];
#pragma unroll 2
    for (int kk = 0; kk < kDFeat / 4; ++kk) {
      const int kb = kk * 4 + half * 2;
      v2f a; a.x = fb[kb]; a.y = fb[kb + 1];       // one A feeds 4 WMMAs
#pragma unroll
      for (int nt = 0; nt < 4; ++nt) {
        v2f b;
        b.x = sW1[kb * kW1S + nt * 16 + lr];
        b.y = sW1[(kb + 1) * kW1S + nt * 16 + lr];
        acc[tm][nt] = __builtin_amdgcn_wmma_f32_16x16x4_f32(
            false, a, false, b, (short)0, acc[tm][nt], false, false);
      }
    }
  }
  __syncthreads();   // everyone done reading feats from sF

  // bias + relu, store hidden h back into sF (cols 0..63)
#pragma unroll
  for (int tm = 0; tm < 2; ++tm) {
    const int mt = wave * 2 + tm;
#pragma unroll
    for (int nt = 0; nt < 4; ++nt) {
      const float bias = sB1[nt * 16 + lr];
#pragma unroll
      for (int r = 0; r < 8; ++r) {
        const int m = mt * 16 + half * 8 + r;
        sF[m * kFS + nt * 16 + lr] = fmaxf(acc[tm][nt][r] + bias, 0.f);
      }
    }
  }
  __syncthreads();

  // Layer 2: h(128x64) @ W2(64x16). Wave owns M-tiles {2w, 2w+1}.
#pragma unroll
  for (int tm = 0; tm < 2; ++tm) {
    const int mt = wave * 2 + tm;
    const float* hb = &sF[(mt * 16 + lr) * kFS];
    v8f acc2 = {};
#pragma unroll 4
    for (int kk = 0; kk < kDHid / 4; ++kk) {
      const int kb = kk * 4 + half * 2;
      v2f a; a.x = hb[kb]; a.y = hb[kb + 1];
      v2f b;
      b.x = sW2[kb * kW2S + lr];
      b.y = sW2[(kb + 1) * kW2S + lr];
      acc2 = __builtin_amdgcn_wmma_f32_16x16x4_f32(
          false, a, false, b, (short)0, acc2, false, false);
    }
    const float bias = sB2[lr];
#pragma unroll
    for (int r = 0; r < 8; ++r) {
      const int m = mt * 16 + half * 8 + r;
      const int pg = blockIdx.x * kBlk + m;
      if (pg < n) out[(size_t)pg * kDOut + lr] = acc2[r] + bias;
    }
  }
}

extern "C" void kernel_launch(void* const* d_in, const int* in_sizes, int n_in,
                              void* d_out, int out_size, void* d_ws, size_t ws_size,
                              hipStream_t stream) {
  const float* in = (const float*)d_in[0];
  const float* t0 = (const float*)d_in[1];
  const float* t1 = (const float*)d_in[2];
  const float* t2 = (const float*)d_in[3];
  const float* W1 = (const float*)d_in[4];
  const float* b1 = (const float*)d_in[5];
  const float* W2 = (const float*)d_in[6];
  const float* b2 = (const float*)d_in[7];
  float* out = (float*)d_out;
  const int n = in_sizes[0] / 4;

  // Per-level resolutions, bit-matching numpy's f64 math:
  // SCALE = exp((log(2048)-log(8))/15) (= 2^(8/15)); res_l = floor(8*SCALE**l)
  ResArr ra;
  const double scale = exp((log(2048.0) - log(8.0)) / 15.0);
  for (int l = 0; l < kL; ++l)
    ra.r[l] = (int)floor(8.0 * pow(scale, (double)l));

  const int blocks = (n + kBlk - 1) / kBlk;
  hipLaunchKernelGGL(hashgrid_mlp, dim3(blocks), dim3(kBlk), 0, stream,
                     in, t0, t1, t2, W1, b1, W2, b2, out, n, ra);
}